// InterpolantActivation_25142738551273
// MI455X (gfx1250) — compile-verified
//
#include <hip/hip_runtime.h>
#include <stdint.h>

// Uniform grid: xg = linspace(-5, 5, 50); h = 10/49; 1/h = 4.9 exactly.
#define XMIN   (-5.0f)
#define INV_H  (4.9f)

// Native clang vector type: accepted by __builtin_nontemporal_{load,store},
// bit-identical layout to HIP float4 (16 bytes).
typedef float v4f __attribute__((ext_vector_type(4)));

// Single linear-interp with linear extrapolation against one activation row
// staged in LDS as (y[k], y[k+1]) pairs. Matches searchsorted-left + clip(0,48).
__device__ __forceinline__ float interp_one(float v, const float2* __restrict__ prow) {
    float t  = (v - XMIN) * INV_H;             // grid-space coordinate
    float tf = floorf(t);
    tf = fminf(fmaxf(tf, 0.0f), 48.0f);        // clamp interval index (extrapolates at ends)
    float2 p = prow[(int)tf];                  // single ds_load_b64
    return __builtin_fmaf(p.y - p.x, t - tf, p.x);
}

__device__ __forceinline__ v4f interp_vec4(v4f v, const float2* __restrict__ prow,
                                           float oth) {
    v4f r;
    r.x = interp_one(v.x, prow) + oth;
    r.y = interp_one(v.y, prow) + oth;
    r.z = interp_one(v.z, prow) + oth;
    r.w = interp_one(v.w, prow) + oth;
    return r;
}

__global__ __launch_bounds__(256) void InterpolantActivation_kernel(
    const float* __restrict__ x,
    const float* __restrict__ act,    // n_act * 50 floats
    float* __restrict__ out,
    long long n4,                     // number of float4 groups
    int n_act,
    int seg_shift)                    // log2((L/n_act)/4); seg = (g & 1023) >> seg_shift
{
    __shared__ __align__(16) float lutRaw[4 * 50];   // raw LUT (async-loaded)
    __shared__ float2 pairs[4 * 49];                 // (y[k], y[k+1]) pairs, 8B aligned
    __shared__ float  zval[4];                       // interp_i(0)
    __shared__ float  otherv[4];                     // sum_{i!=j} interp_i(0)

    const int tid = threadIdx.x;

    // ---- Stage LUT into LDS via CDNA5 async global->LDS DMA (ASYNCcnt path) ----
    if (tid < 32) {
        const int ndw = n_act * 50;                           // dwords to move (<=200)
        uint32_t ldsBase = (uint32_t)(uintptr_t)(&lutRaw[0]); // flat->LDS: low 32 bits
        uint64_t gBase   = (uint64_t)(uintptr_t)act;
        for (int k = tid; k < ndw; k += 32) {
            uint32_t la = ldsBase + (uint32_t)k * 4u;
            uint64_t ga = gBase + (uint64_t)k * 4u;
            asm volatile("global_load_async_to_lds_b32 %0, %1, off"
                         :: "v"(la), "v"(ga) : "memory");
        }
        asm volatile("s_wait_asynccnt 0x0" ::: "memory");
    }
    __syncthreads();

    // ---- Build pair table + per-activation constants interp_i(0) ----
    const int npairs = n_act * 49;
    for (int e = tid; e < npairs; e += (int)blockDim.x) {
        int i = e / 49;
        int k = e - i * 49;
        float a = lutRaw[i * 50 + k];
        float b = lutRaw[i * 50 + k + 1];
        pairs[e] = make_float2(a, b);
    }
    if (tid < n_act) {
        // t0 = (0 - (-5)) * 4.9 = 24.5 -> interval 24, fraction exactly 0.5
        float y0 = lutRaw[tid * 50 + 24];
        float y1 = lutRaw[tid * 50 + 25];
        zval[tid] = y0 + (y1 - y0) * 0.5f;
    }
    __syncthreads();
    if (tid == 0) {
        float s = 0.0f;
        for (int i = 0; i < n_act; ++i) s += zval[i];
        for (int j = 0; j < n_act; ++j) otherv[j] = s - zval[j];
    }
    __syncthreads();

    // ---- Streaming pass: NT b128 in, NT b128 out, one ds_load_b64 per element ----
    // x is read once and out written once (never re-read): non-temporal hints keep
    // the 256 MB of streaming traffic from evicting the (reused) LUT in L2.
    const v4f* __restrict__ xv = (const v4f*)x;
    v4f* __restrict__ ov = (v4f*)out;
    const long long stride = (long long)gridDim.x * (long long)blockDim.x;
    const long long g0 = (long long)blockIdx.x * blockDim.x + tid;

    // Unroll-by-2 main body: two b128 loads in flight per wave iteration.
    long long g = g0;
    for (; g + stride < n4; g += 2 * stride) {
        long long ga = g;
        long long gb = g + stride;

        int segA = ((int)(ga & 1023LL)) >> seg_shift;
        int segB = ((int)(gb & 1023LL)) >> seg_shift;

        v4f va = __builtin_nontemporal_load(xv + ga);
        v4f vb = __builtin_nontemporal_load(xv + gb);
        __builtin_prefetch(xv + ga + 2 * stride, 0, 0);   // global_prefetch next wave of data

        v4f ra = interp_vec4(va, &pairs[segA * 49], otherv[segA]);
        v4f rb = interp_vec4(vb, &pairs[segB * 49], otherv[segB]);

        __builtin_nontemporal_store(ra, ov + ga);
        __builtin_nontemporal_store(rb, ov + gb);
    }
    // Remainder (at most one group per thread with the exact-coverage launch).
    for (; g < n4; g += stride) {
        int seg = ((int)(g & 1023LL)) >> seg_shift;
        v4f v = __builtin_nontemporal_load(xv + g);
        v4f r = interp_vec4(v, &pairs[seg * 49], otherv[seg]);
        __builtin_nontemporal_store(r, ov + g);
    }
}

extern "C" void kernel_launch(void* const* d_in, const int* in_sizes, int n_in,
                              void* d_out, int out_size, void* d_ws, size_t ws_size,
                              hipStream_t stream) {
    (void)n_in; (void)d_ws; (void)ws_size; (void)out_size;
    const float* x   = (const float*)d_in[0];   // [8192, 4096] fp32
    const float* act = (const float*)d_in[1];   // [n_act, 50] fp32
    float* out = (float*)d_out;                 // [8192, 4096] fp32

    long long n = (long long)in_sizes[0];
    long long n4 = n / 4;

    int n_act = in_sizes[1] / 50;
    if (n_act < 1) n_act = 1;
    if (n_act > 4) n_act = 4;

    const int L = 4096;                 // row length per reference
    int split4 = (L / n_act) / 4;       // float4-groups per segment (power of two)
    int seg_shift = 0;
    while ((1 << (seg_shift + 1)) <= split4) ++seg_shift;

    const int threads = 256;            // 8 waves per block (wave32)
    const int blocks  = 4096;           // 4096*256 threads -> exactly 8 float4s each

    InterpolantActivation_kernel<<<dim3(blocks), dim3(threads), 0, stream>>>(
        x, act, out, n4, n_act, seg_shift);
}